// LocallyConnected1d_14551349199055
// MI455X (gfx1250) — compile-verified
//
#include <hip/hip_runtime.h>

// LocallyConnected1d: out[b,n,o] = sum_{c,t} x[b,c,o+t] * w[n,c,o,t]
// Per-o GEMM (M=n=32, N=b=32, K=c*t=224) via v_wmma_f32_16x16x4_f32.
// Memory-bound (~302 MB, weight 235 MB) => fp32 WMMA, 8 consecutive o's per
// workgroup (1 per wave) so weight lines are shared inside the WGP, LDS
// transpose epilogue for contiguous b64 output stores.

typedef float v2f __attribute__((ext_vector_type(2)));
typedef float v8f __attribute__((ext_vector_type(8)));

#define BATCH 32
#define CIN   32
#define COUT  32
#define ISZ   8192
#define OSZ   8186
#define KW    7
#define O_PER_BLOCK 8   // 8 waves * 1 o each

__global__ __launch_bounds__(256)
void lc1d_wmma_kernel(const float* __restrict__ x,
                      const float* __restrict__ w,
                      float* __restrict__ out) {
  __shared__ float sout[BATCH * COUT * O_PER_BLOCK];  // 32 KB transpose buffer

  const int wave = threadIdx.x >> 5;
  const int lane = threadIdx.x & 31;
  const int row  = lane & 15;   // A row (n) / B column (b)
  const int half = lane >> 4;   // selects K pair in A/B, M+8 in D
  const int oRaw = blockIdx.x * O_PER_BLOCK + wave;
  // No early return (barrier below). Clamp keeps every address in-bounds and
  // EXEC all-1s for WMMA; clamped waves' results are never copied to global.
  const int o = (oRaw < OSZ) ? oRaw : (OSZ - 1);

  // Per-lane K offsets, invariant across channels. K padded 7 -> 8.
  const int  kA   = half * 2;              // chunk 0: elements kA, kA+1
  const int  kB   = 4 + half * 2;          // chunk 1: first element (4 or 6)
  const bool tail = (half != 0);           // chunk 1, k==6: k+1 is the pad
  const int  kBhi = tail ? kB : kB + 1;    // clamped in-bounds second element

  // 2x2 tile of 16x16 f32 accumulators: [n-half][b-half]
  v8f acc00 = {}; v8f acc01 = {}; v8f acc10 = {}; v8f acc11 = {};

  const long wNH  = (long)16 * CIN * OSZ * KW;  // n += 16 stride in weight
  const long xBH  = (long)16 * CIN * ISZ;       // b += 16 stride in x
  const long wCST = (long)OSZ * KW;             // c += 1 stride in weight
  long wIdx = ((long)row * CIN * OSZ + o) * KW; // A base: n = row
  long xIdx = (long)row * CIN * ISZ + o;        // B base: b = row

  for (int c = 0; c < CIN; ++c) {
    // Pull next channel's contiguous weight run WGP-scope (shared by 8 waves)
    __builtin_prefetch(w + wIdx + wCST, 0, 3);

    // ---- chunk 0: K = 0..3 (adjacent dwords -> merged global_load_b64) ----
    {
      v2f a0, a1, b0, b1;
      a0[0] = w[wIdx + kA];        a0[1] = w[wIdx + kA + 1];
      a1[0] = w[wIdx + wNH + kA];  a1[1] = w[wIdx + wNH + kA + 1];
      b0[0] = x[xIdx + kA];        b0[1] = x[xIdx + kA + 1];
      b1[0] = x[xIdx + xBH + kA];  b1[1] = x[xIdx + xBH + kA + 1];
      acc00 = __builtin_amdgcn_wmma_f32_16x16x4_f32(false, a0, false, b0,
                                                    (short)0, acc00, false, false);
      acc01 = __builtin_amdgcn_wmma_f32_16x16x4_f32(false, a0, false, b1,
                                                    (short)0, acc01, false, false);
      acc10 = __builtin_amdgcn_wmma_f32_16x16x4_f32(false, a1, false, b0,
                                                    (short)0, acc10, false, false);
      acc11 = __builtin_amdgcn_wmma_f32_16x16x4_f32(false, a1, false, b1,
                                                    (short)0, acc11, false, false);
    }

    // ---- chunk 1: K = 4..7 (k=7 is zero-pad on the A side only) ----
    {
      // Unconditional loads from clamped addresses, then v_cndmask to zero:
      // no EXEC save/restore branches.
      const float ah0 = w[wIdx + kBhi];
      const float ah1 = w[wIdx + wNH + kBhi];
      v2f a0, a1, b0, b1;
      a0[0] = w[wIdx + kB];        a0[1] = tail ? 0.0f : ah0;
      a1[0] = w[wIdx + wNH + kB];  a1[1] = tail ? 0.0f : ah1;
      // B's k=7 value multiplies A's zero -> don't-care, just keep in-bounds.
      b0[0] = x[xIdx + kB];        b0[1] = x[xIdx + kBhi];
      b1[0] = x[xIdx + xBH + kB];  b1[1] = x[xIdx + xBH + kBhi];
      acc00 = __builtin_amdgcn_wmma_f32_16x16x4_f32(false, a0, false, b0,
                                                    (short)0, acc00, false, false);
      acc01 = __builtin_amdgcn_wmma_f32_16x16x4_f32(false, a0, false, b1,
                                                    (short)0, acc01, false, false);
      acc10 = __builtin_amdgcn_wmma_f32_16x16x4_f32(false, a1, false, b0,
                                                    (short)0, acc10, false, false);
      acc11 = __builtin_amdgcn_wmma_f32_16x16x4_f32(false, a1, false, b1,
                                                    (short)0, acc11, false, false);
    }

    wIdx += wCST;
    xIdx += ISZ;
  }

  // D layout (16x16 f32): VGPR r -> M = r + half*8, N = row. M = n, N = b.
  // Stage into LDS so output becomes contiguous runs of 8 o's per (b,n) row.
  #pragma unroll
  for (int r = 0; r < 8; ++r) {
    const int n = half * 8 + r;
    const int b = row;
    sout[((b     ) * COUT + n     ) * O_PER_BLOCK + wave] = acc00[r];
    sout[((b + 16) * COUT + n     ) * O_PER_BLOCK + wave] = acc01[r];
    sout[((b     ) * COUT + n + 16) * O_PER_BLOCK + wave] = acc10[r];
    sout[((b + 16) * COUT + n + 16) * O_PER_BLOCK + wave] = acc11[r];
  }
  __syncthreads();

  // 1024 rows (b*32+n), 8 floats each; 256 threads -> 4 rows per thread.
  const int o0  = blockIdx.x * O_PER_BLOCK;
  const int tid = threadIdx.x;
  if (o0 + O_PER_BLOCK <= OSZ) {
    #pragma unroll
    for (int rr = 0; rr < 4; ++rr) {
      const int rowid = rr * 256 + tid;
      const float2* src = (const float2*)&sout[rowid * O_PER_BLOCK];
      float2* dst = (float2*)&out[(long)rowid * OSZ + o0];  // 8B aligned
      #pragma unroll
      for (int j = 0; j < 4; ++j) dst[j] = src[j];          // global_store_b64
    }
  } else {
    // Last block: per-element guard filters out the clamped waves' columns.
    for (int rr = 0; rr < 4; ++rr) {
      const int rowid = rr * 256 + tid;
      for (int j = 0; j < O_PER_BLOCK; ++j)
        if (o0 + j < OSZ)
          out[(long)rowid * OSZ + o0 + j] = sout[rowid * O_PER_BLOCK + j];
    }
  }
}

extern "C" void kernel_launch(void* const* d_in, const int* in_sizes, int n_in,
                              void* d_out, int out_size, void* d_ws, size_t ws_size,
                              hipStream_t stream) {
  const float* x = (const float*)d_in[0];  // (32, 32, 8192)
  const float* w = (const float*)d_in[1];  // (1, 32, 32, 8186, 7) flat
  float* out = (float*)d_out;              // (32, 32, 8186) flat

  dim3 grid((OSZ + O_PER_BLOCK - 1) / O_PER_BLOCK);  // 1024 blocks
  dim3 block(256);                                   // 8 wave32 waves
  hipLaunchKernelGGL(lc1d_wmma_kernel, grid, block, 0, stream, x, w, out);
}